// SortAndSelectNeighbours_66460323938758
// MI455X (gfx1250) — compile-verified
//
#include <hip/hip_runtime.h>
#include <stdint.h>

#define KIN  128
#define KOUT 64
#define ROWS_PER_BLOCK 8
#define RADIUS 0.5f
#define BIGF   1.0e9f

__device__ __forceinline__ uint64_t shflxor64(uint64_t v, int m) {
    uint32_t lo = (uint32_t)v;
    uint32_t hi = (uint32_t)(v >> 32);
    lo = __shfl_xor(lo, m, 32);
    hi = __shfl_xor(hi, m, 32);
    return ((uint64_t)hi << 32) | (uint64_t)lo;
}

__device__ __forceinline__ void cswap(uint64_t& a, uint64_t& b, bool asc) {
    uint64_t x = a, y = b;
    bool lt = x < y;
    uint64_t mn = lt ? x : y;
    uint64_t mx = lt ? y : x;
    a = asc ? mn : mx;
    b = asc ? mx : mn;
}

__global__ __launch_bounds__(256)
void SortAndSelectNeighbours_kernel(const float* __restrict__ dist,
                                    const int*   __restrict__ nidx,
                                    float*       __restrict__ outd,
                                    int*         __restrict__ outi,
                                    int n)
{
    __shared__ __align__(16) float sd[ROWS_PER_BLOCK][KIN];
    __shared__ __align__(16) int   si[ROWS_PER_BLOCK][KIN];

    const int lane = threadIdx.x & 31;
    const int w    = threadIdx.x >> 5;
    const int row  = blockIdx.x * ROWS_PER_BLOCK + w;
    if (row >= n) return;

    // ---- Stage this wave's row into LDS via CDNA5 async global->LDS DMA ----
    {
        const float* gd = dist + (size_t)row * KIN + lane * 4;
        const int*   gi = nidx + (size_t)row * KIN + lane * 4;
        uint32_t ld = (uint32_t)(uintptr_t)(&sd[w][lane * 4]);
        uint32_t li = (uint32_t)(uintptr_t)(&si[w][lane * 4]);
        asm volatile("global_load_async_to_lds_b128 %0, %1, off"
                     :: "v"(ld), "v"(gd) : "memory");
        asm volatile("global_load_async_to_lds_b128 %0, %1, off"
                     :: "v"(li), "v"(gi) : "memory");
        asm volatile("s_wait_asynccnt 0x0" ::: "memory");
    }

    // ---- Build 64-bit sort keys: (masked_dist_bits << 32) | column ----
    // Positive floats compare correctly as unsigned ints; embedding the
    // column in the low bits makes the sort stable (matches jnp.argsort).
    float4 d4 = *reinterpret_cast<const float4*>(&sd[w][lane * 4]);
    int4   i4 = *reinterpret_cast<const int4*>(&si[w][lane * 4]);

    uint64_t key[4];
    {
        const float dv[4] = {d4.x, d4.y, d4.z, d4.w};
        const int   iv[4] = {i4.x, i4.y, i4.z, i4.w};
#pragma unroll
        for (int r = 0; r < 4; ++r) {
            float m = (iv[r] < 0) ? BIGF : dv[r];
            uint32_t mb = __float_as_uint(m);
            key[r] = ((uint64_t)mb << 32) | (uint32_t)(lane * 4 + r);
        }
    }

    // ---- Wave-level bitonic sort of 128 keys, element e = lane*4 + r ----
    // j in {1,2}: in-register; j >= 4: cross-lane shuffle-xor (lm = j/4).
    cswap(key[0], key[1], true);    // k = 2, pair (e, e^1): e&2==0 -> asc
    cswap(key[2], key[3], false);   //                       e&2!=0 -> desc

#pragma unroll
    for (int kk = 4; kk <= 128; kk <<= 1) {
        const bool asc = (((lane * 4) & kk) == 0); // kk>=4: lane-uniform
#pragma unroll
        for (int j = kk >> 1; j >= 4; j >>= 1) {
            const int  lm      = j >> 2;
            const bool lower   = ((lane & lm) == 0);
            const bool takemin = (lower == asc);
#pragma unroll
            for (int r = 0; r < 4; ++r) {
                uint64_t o  = shflxor64(key[r], lm);
                uint64_t mn = key[r] < o ? key[r] : o;
                uint64_t mx = key[r] < o ? o      : key[r];
                key[r] = takemin ? mn : mx;
            }
        }
        // j = 2 then j = 1 (in-register)
        cswap(key[0], key[2], asc);
        cswap(key[1], key[3], asc);
        cswap(key[0], key[1], asc);
        cswap(key[2], key[3], asc);
    }

    // ---- Top-64 lives in lanes 0..15 (positions p = lane*4 + r) ----
    if (lane < 16) {
        float od[4];
        int   oi[4];
#pragma unroll
        for (int r = 0; r < 4; ++r) {
            uint32_t col = (uint32_t)key[r] & (KIN - 1);  // original column
            float dv = sd[w][col];   // ORIGINAL distance (unmasked)
            int   iv = si[w][col];
            bool beyond = dv > RADIUS;
            od[r] = beyond ? 0.0f : dv;
            oi[r] = beyond ? -1   : iv;   // iv==-1 (invalid) stays -1
        }
        size_t base = (size_t)row * KOUT + (size_t)lane * 4;
        *reinterpret_cast<float4*>(outd + base) = make_float4(od[0], od[1], od[2], od[3]);
        *reinterpret_cast<int4*>(outi + base)   = make_int4(oi[0], oi[1], oi[2], oi[3]);
    }
}

extern "C" void kernel_launch(void* const* d_in, const int* in_sizes, int n_in,
                              void* d_out, int out_size, void* d_ws, size_t ws_size,
                              hipStream_t stream) {
    const float* dist = (const float*)d_in[0];
    const int*   nid  = (const int*)d_in[1];
    const int n = in_sizes[0] / KIN;             // 400000

    float* outd = (float*)d_out;                 // sdist [N,64] f32
    int*   outi = (int*)(outd + (size_t)n * KOUT); // snidx [N,64] i32 (bit-cast half)

    const int blocks = (n + ROWS_PER_BLOCK - 1) / ROWS_PER_BLOCK;
    SortAndSelectNeighbours_kernel<<<blocks, 256, 0, stream>>>(dist, nid, outd, outi, n);
}